// NCDETransformer_49426483642796
// MI455X (gfx1250) — compile-verified
//
#include <hip/hip_runtime.h>
#include <hip/hip_bf16.h>

typedef float v2f __attribute__((ext_vector_type(2)));
typedef float v8f __attribute__((ext_vector_type(8)));
typedef unsigned int u32x4 __attribute__((ext_vector_type(4)));
typedef unsigned int u32x8 __attribute__((ext_vector_type(8)));

#define B_    2
#define S_    40
#define DM    256
#define H_    8
#define DK    32
#define HB    16      // H_*B_
#define NSEG  32      // 8 chunks * 4 segments
#define NIV   39      // spline intervals (S_-1)
#define INV_SQRT_DK 0.17677669529663687f

// ---------------------------------------------------------------------------
// K1: fused q/k/v projections.  (80 x 256) @ (256 x 256) via WMMA f32 16x16x4,
// scattered into (HB, S, DK) head-major layout.
// grid = 3*5*16 = 240 blocks of 1 wave (32 threads).
// ---------------------------------------------------------------------------
__global__ void proj_kernel(const float* __restrict__ q, const float* __restrict__ k,
                            const float* __restrict__ v,
                            const float* __restrict__ Wq, const float* __restrict__ Wk,
                            const float* __restrict__ Wv,
                            float* __restrict__ qh, float* __restrict__ kh,
                            float* __restrict__ vh)
{
    int bi    = blockIdx.x;
    int which = bi / 80;              // 0=q 1=k 2=v
    int tile  = bi % 80;
    int mt = tile % 5, nt = tile / 5; // M=80 -> 5 tiles, N=256 -> 16 tiles
    const float* A = (which == 0) ? q  : (which == 1) ? k  : v;
    const float* W = (which == 0) ? Wq : (which == 1) ? Wk : Wv;
    float*       O = (which == 0) ? qh : (which == 1) ? kh : vh;

    int lane  = threadIdx.x;
    int m     = lane & 15;            // M row / N col within tile
    int khalf = lane >> 4;            // K half select (lanes 16-31 hold K+2)

    v8f acc = {0.f, 0.f, 0.f, 0.f, 0.f, 0.f, 0.f, 0.f};
    for (int kb = 0; kb < DM; kb += 4) {
        int ka = kb + 2 * khalf;
        v2f a, b;
        a.x = A[(mt * 16 + m) * DM + ka];
        a.y = A[(mt * 16 + m) * DM + ka + 1];
        b.x = W[ka * DM + nt * 16 + m];
        b.y = W[(ka + 1) * DM + nt * 16 + m];
        acc = __builtin_amdgcn_wmma_f32_16x16x4_f32(false, a, false, b,
                                                    (short)0, acc, false, false);
    }
#pragma unroll
    for (int r = 0; r < 8; ++r) {
        int row = mt * 16 + khalf * 8 + r;   // b*S+s
        int col = nt * 16 + m;               // h*DK+c
        int b_ = row / S_, s = row % S_;
        int h  = col >> 5, c = col & 31;
        O[((h * B_ + b_) * S_ + s) * DK + c] = acc[r];
    }
}

// ---------------------------------------------------------------------------
// K2: natural cubic spline fit on qh over time.  One thread per (hb, channel)
// = 512 channels; Thomas algorithm on the 38-unknown tridiagonal system.
// ---------------------------------------------------------------------------
__global__ void spline_kernel(const float* __restrict__ qh, const float* __restrict__ times,
                              float* __restrict__ aC, float* __restrict__ bC,
                              float* __restrict__ cC, float* __restrict__ dC)
{
    int tid = threadIdx.x;            // 512
    int hb = tid >> 5, c = tid & 31;

    float t[S_], y[S_], hseg[NIV], slope[NIV], rhs[38], cp[38], dp[38], mm[S_];
    for (int i = 0; i < S_; ++i) t[i] = times[i];               // row 0 of (B,S)
    for (int i = 0; i < S_; ++i) y[i] = qh[(hb * S_ + i) * DK + c];
    for (int i = 0; i < NIV; ++i) {
        hseg[i]  = t[i + 1] - t[i];
        slope[i] = (y[i + 1] - y[i]) / hseg[i];
    }
    for (int i = 0; i < 38; ++i) rhs[i] = 6.f * (slope[i + 1] - slope[i]);

    float main0 = 2.f * (hseg[0] + hseg[1]);
    cp[0] = hseg[1] / main0;
    dp[0] = rhs[0] / main0;
    for (int i = 1; i < 38; ++i) {
        float den = 2.f * (hseg[i] + hseg[i + 1]) - hseg[i] * cp[i - 1];
        cp[i] = ((i < 37) ? hseg[i + 1] : 0.f) / den;
        dp[i] = (rhs[i] - hseg[i] * dp[i - 1]) / den;
    }
    mm[0] = 0.f; mm[39] = 0.f;
    mm[38] = dp[37];
    for (int i = 36; i >= 0; --i) mm[i + 1] = dp[i] - cp[i] * mm[i + 2];

    for (int i = 0; i < NIV; ++i) {
        int o = (hb * NIV + i) * DK + c;
        aC[o] = y[i];
        bC[o] = slope[i] - hseg[i] * (2.f * mm[i] + mm[i + 1]) * (1.f / 6.f);
        cC[o] = 0.5f * mm[i];
        dC[o] = (mm[i + 1] - mm[i]) / (6.f * hseg[i]);
    }
}

// ---------------------------------------------------------------------------
// (48x32) = (48x32)@(32x32) tile GEMM via WMMA f32 16x16x4. 6 waves, 1 tile ea.
// ---------------------------------------------------------------------------
__device__ __forceinline__ void gemm_48x32x32(const float* __restrict__ A,
                                              const float* __restrict__ Bm,
                                              float* __restrict__ C,
                                              const float* __restrict__ bias,
                                              bool do_tanh, int tid)
{
    int w = tid >> 5;
    if (w < 6) {
        int lane = tid & 31;
        int m = lane & 15, khalf = lane >> 4;
        int mt = w >> 1, nt = w & 1;
        v8f acc = {0.f, 0.f, 0.f, 0.f, 0.f, 0.f, 0.f, 0.f};
#pragma unroll
        for (int kb = 0; kb < 32; kb += 4) {
            int ka = kb + 2 * khalf;
            v2f a, b;
            a.x = A[(mt * 16 + m) * 32 + ka];
            a.y = A[(mt * 16 + m) * 32 + ka + 1];
            b.x = Bm[ka * 32 + nt * 16 + m];
            b.y = Bm[(ka + 1) * 32 + nt * 16 + m];
            acc = __builtin_amdgcn_wmma_f32_16x16x4_f32(false, a, false, b,
                                                        (short)0, acc, false, false);
        }
#pragma unroll
        for (int r = 0; r < 8; ++r) {
            int row = mt * 16 + khalf * 8 + r;
            int col = nt * 16 + m;
            float val = acc[r] + bias[col];
            C[row * 32 + col] = do_tanh ? tanhf(val) : val;
        }
    }
}

// ---------------------------------------------------------------------------
// K3: the Neural-CDE integration.  One workgroup per (segment, hb) tile
// (512 blocks x 256 threads).  Wk2 (128 KB) pulled into LDS once via the
// Tensor Data Mover; per-stage rhs = tanh(K@Wk1+bk1) @ G(dq) + bterm with
// G[i,h] = sum_c Wk2[i,h*32+c]*dq[c]  (algebraic contraction of the einsum).
// ---------------------------------------------------------------------------
__global__ void ode_kernel(const float* __restrict__ kh, const float* __restrict__ times,
                           const float* __restrict__ aC, const float* __restrict__ bC,
                           const float* __restrict__ cC, const float* __restrict__ dC,
                           const float* __restrict__ Wk1, const float* __restrict__ bk1,
                           const float* __restrict__ Wk2, const float* __restrict__ bk2,
                           float* __restrict__ segScores)
{
    extern __shared__ float smem[];
    float* sWk2 = smem;                 // 32*1024
    float* sWk1 = sWk2 + 32 * 1024;     // 1024
    float* sbk1 = sWk1 + 1024;          // 32
    float* sbk2 = sbk1 + 32;            // 1024
    float* sKst = sbk2 + 1024;          // 48*32 state (rows 40..47 zero-padded)
    float* sE   = sKst + 1536;          // eval point
    float* sK1  = sE + 1536;
    float* sK2  = sK1 + 1536;
    float* sK3  = sK2 + 1536;
    float* sHdn = sK3 + 1536;           // 48*32
    float* sG   = sHdn + 1536;          // 32*32
    float* sBt  = sG + 1024;            // 32
    float* sDq  = sBt + 32;             // 32
    float* sQt  = sDq + 32;             // 32
    float* sT   = sQt + 32;             // 40

    int tid = threadIdx.x;
    int blk = blockIdx.x;
    int seg = blk & 31;
    int hb  = blk >> 5;
    int chunk = seg >> 2, j = seg & 3;

    // ---- TDM: async copy Wk2 (2-D tensor: 32 rows x 1024 f32) into LDS ----
    if (tid == 0) {
        unsigned long long ga = (unsigned long long)(const void*)Wk2;
        unsigned ldsoff = (unsigned)(unsigned long long)(void*)sWk2; // low 32b of flat = LDS addr
        u32x4 g0;
        g0[0] = 1u;                                  // count=1, no gather
        g0[1] = ldsoff;                              // lds_addr
        g0[2] = (unsigned)ga;                        // global_addr[31:0]
        g0[3] = ((unsigned)(ga >> 32) & 0x01FFFFFFu) // global_addr[56:32]
                | 0x80000000u;                       // type=2 ("image")
        u32x8 g1;
        g1[0] = 0x00020000u;                         // wg_mask=0, data_size=4B
        g1[1] = (1024u & 0xFFFFu) << 16;             // tensor_dim0[15:0]=1024
        g1[2] = (32u << 16);                         // dim0 hi=0, tensor_dim1[15:0]=32
        g1[3] = (1024u << 16);                       // dim1 hi=0, tile_dim0=1024
        g1[4] = 32u;                                 // tile_dim1=32, tile_dim2=0
        g1[5] = 1024u;                               // tensor_dim0_stride lo
        g1[6] = 0u;
        g1[7] = 0u;
        asm volatile("tensor_load_to_lds %0, %1" :: "s"(g0), "s"(g1) : "memory");
    }

    // cooperative loads of the small operands while the TDM streams Wk2
    for (int e = tid; e < 1024; e += 256) { sWk1[e] = Wk1[e]; sbk2[e] = bk2[e]; }
    if (tid < 32) sbk1[tid] = bk1[tid];
    if (tid < S_) sT[tid] = times[tid];
    for (int e = tid; e < 1536; e += 256)
        sKst[e] = (e < 1280) ? kh[hb * 1280 + e] : 0.f;   // k_init tile (40x32)

    if (tid == 0) __builtin_amdgcn_s_wait_tensorcnt(0);
    __syncthreads();

    float tlo = sT[chunk * 5], thi = sT[chunk * 5 + 4];
    float t0 = tlo + (thi - tlo) * 0.25f * (float)j;
    float t1 = tlo + (thi - tlo) * 0.25f * (float)(j + 1);
    float dt = (t1 - t0) * 0.25f;                 // / M_STEPS
    const float c13 = 1.f / 3.f;

    auto rhs = [&](const float* Ein, float* Out, float ts) {
        if (tid < 32) {                           // dq = spline_deriv(ts) for this hb
            int idx = 0;
            for (int i = 1; i < S_; ++i) if (sT[i] <= ts) idx = i;
            if (idx > 38) idx = 38;
            float f = ts - sT[idx];
            int base = (hb * NIV + idx) * DK + tid;
            sDq[tid] = bC[base] + f * (2.f * cC[base] + 3.f * f * dC[base]);
        }
        gemm_48x32x32(Ein, sWk1, sHdn, sbk1, true, tid);   // hdn = tanh(E@Wk1+bk1)
        __syncthreads();
        for (int e = tid; e < 1024; e += 256) {            // G[i,h] = Wk2[i,h,:].dq
            int i = e >> 5, h2 = e & 31;
            const float* wr = &sWk2[i * 1024 + h2 * 32];
            float sum = 0.f;
#pragma unroll
            for (int c = 0; c < 32; ++c) sum += wr[c] * sDq[c];
            sG[i * 32 + h2] = sum;
        }
        if (tid < 32) {                                    // bterm[h] = bk2[h,:].dq
            float bt = 0.f;
#pragma unroll
            for (int c = 0; c < 32; ++c) bt += sbk2[tid * 32 + c] * sDq[c];
            sBt[tid] = bt;
        }
        __syncthreads();
        gemm_48x32x32(sHdn, sG, Out, sBt, false, tid);     // Out = hdn@G + bterm
        __syncthreads();
    };

    for (int ms = 0; ms < 4; ++ms) {                       // M_STEPS, RK4-variant
        float ts = t0 + dt * (float)ms;
        rhs(sKst, sK1, ts);
        for (int e = tid; e < 1536; e += 256) sE[e] = sKst[e] + dt * c13 * sK1[e];
        __syncthreads();
        rhs(sE, sK2, ts + dt * c13);
        for (int e = tid; e < 1536; e += 256) sE[e] = sKst[e] + dt * (sK2[e] - c13 * sK1[e]);
        __syncthreads();
        rhs(sE, sK3, ts + 2.f * dt * c13);
        for (int e = tid; e < 1536; e += 256) sE[e] = sKst[e] + dt * (sK1[e] - sK2[e] + sK3[e]);
        __syncthreads();
        rhs(sE, sE, ts + dt);                              // k4 -> sE
        for (int e = tid; e < 1536; e += 256)
            sKst[e] += dt * 0.125f * (sK1[e] + 3.f * (sK2[e] + sK3[e]) + sE[e]);
        __syncthreads();
    }

    if (tid < 32) {                                        // q_t = spline_eval(t1)
        int idx = 0;
        for (int i = 1; i < S_; ++i) if (sT[i] <= t1) idx = i;
        if (idx > 38) idx = 38;
        float f = t1 - sT[idx];
        int base = (hb * NIV + idx) * DK + tid;
        sQt[tid] = aC[base] + f * (bC[base] + f * (cC[base] + f * dC[base]));
    }
    __syncthreads();
    if (tid < S_) {                                        // score[s] = q_t . k_t[s]
        float acc = 0.f;
#pragma unroll
        for (int c = 0; c < 32; ++c) acc += sQt[c] * sKst[tid * 32 + c];
        segScores[(seg * HB + hb) * S_ + tid] = acc * INV_SQRT_DK;
    }
}

// ---------------------------------------------------------------------------
// K4: chunk-mean -> softmax -> attn @ V, scatter to (B*S, H*DV) layout.
// ---------------------------------------------------------------------------
__global__ void attn_kernel(const float* __restrict__ segScores,
                            const float* __restrict__ vh, float* __restrict__ outH)
{
    int hb = blockIdx.x;
    int i  = threadIdx.x;
    if (i >= S_) return;
    int chunk = i / 5;                 // CHUNK repeat
    float sc[S_];
    float mx = -1e30f;
    for (int s = 0; s < S_; ++s) {
        float sum = 0.f;
        for (int jj = 0; jj < 4; ++jj)
            sum += segScores[((chunk * 4 + jj) * HB + hb) * S_ + s];
        float val = sum * 0.25f * INV_SQRT_DK;   // mean, then second 1/sqrt(dk)
        sc[s] = val;
        mx = fmaxf(mx, val);
    }
    float den = 0.f;
    for (int s = 0; s < S_; ++s) { sc[s] = __expf(sc[s] - mx); den += sc[s]; }
    float inv = 1.f / den;
    int h = hb >> 1, b = hb & 1;       // hb = h*B + b
    for (int d = 0; d < DK; ++d) {
        float acc = 0.f;
        for (int s = 0; s < S_; ++s) acc += sc[s] * vh[(hb * S_ + s) * DK + d];
        outH[(b * S_ + i) * DM + h * DK + d] = acc * inv;
    }
}

// ---------------------------------------------------------------------------
// K5: final projection (80 x 256) @ (256 x 256) via WMMA.
// ---------------------------------------------------------------------------
__global__ void fc_kernel(const float* __restrict__ X, const float* __restrict__ W,
                          float* __restrict__ out)
{
    int tile = blockIdx.x;             // 80 tiles
    int mt = tile % 5, nt = tile / 5;
    int lane = threadIdx.x;
    int m = lane & 15, khalf = lane >> 4;
    v8f acc = {0.f, 0.f, 0.f, 0.f, 0.f, 0.f, 0.f, 0.f};
    for (int kb = 0; kb < DM; kb += 4) {
        int ka = kb + 2 * khalf;
        v2f a, b;
        a.x = X[(mt * 16 + m) * DM + ka];
        a.y = X[(mt * 16 + m) * DM + ka + 1];
        b.x = W[ka * DM + nt * 16 + m];
        b.y = W[(ka + 1) * DM + nt * 16 + m];
        acc = __builtin_amdgcn_wmma_f32_16x16x4_f32(false, a, false, b,
                                                    (short)0, acc, false, false);
    }
#pragma unroll
    for (int r = 0; r < 8; ++r)
        out[(mt * 16 + khalf * 8 + r) * DM + nt * 16 + m] = acc[r];
}

// ---------------------------------------------------------------------------
extern "C" void kernel_launch(void* const* d_in, const int* in_sizes, int n_in,
                              void* d_out, int out_size, void* d_ws, size_t ws_size,
                              hipStream_t stream)
{
    (void)in_sizes; (void)n_in; (void)out_size; (void)ws_size;
    const float* q     = (const float*)d_in[0];
    const float* k     = (const float*)d_in[1];
    const float* v     = (const float*)d_in[2];
    const float* times = (const float*)d_in[3];
    const float* Wq    = (const float*)d_in[4];
    const float* Wk    = (const float*)d_in[5];
    const float* Wv    = (const float*)d_in[6];
    const float* Wk1   = (const float*)d_in[7];
    const float* bk1   = (const float*)d_in[8];
    const float* Wk2   = (const float*)d_in[9];
    const float* bk2   = (const float*)d_in[10];
    const float* Wfc   = (const float*)d_in[11];
    float* out = (float*)d_out;
    float* ws  = (float*)d_ws;

    // workspace carve (floats): ~713 KB total
    float* qh   = ws;                       // 16*40*32
    float* kh   = qh + 20480;
    float* vh   = kh + 20480;
    float* aC   = vh + 20480;               // 16*39*32 each
    float* bC   = aC + HB * NIV * DK;
    float* cC   = bC + HB * NIV * DK;
    float* dC   = cC + HB * NIV * DK;
    float* segS = dC + HB * NIV * DK;       // 32*16*40
    float* outH = segS + NSEG * HB * S_;    // 80*256

    proj_kernel<<<240, 32, 0, stream>>>(q, k, v, Wq, Wk, Wv, qh, kh, vh);
    spline_kernel<<<1, 512, 0, stream>>>(qh, times, aC, bC, cC, dC);
    size_t smem = 45224 * sizeof(float);    // ~181 KB dynamic LDS (<= 320 KB/WGP)
    ode_kernel<<<NSEG * HB, 256, smem, stream>>>(kh, times, aC, bC, cC, dC,
                                                 Wk1, bk1, Wk2, bk2, segS);
    attn_kernel<<<HB, 64, 0, stream>>>(segS, vh, outH);
    fc_kernel<<<80, 32, 0, stream>>>(outH, Wfc, out);
}